// FPCELossV3_45251775431144
// MI455X (gfx1250) — compile-verified
//
#include <hip/hip_runtime.h>

// FPCELossV3 for MI455X (gfx1250, wave32).
// Memory-bound streaming reduction: 256MB of pred read once (~11us @ 23.3TB/s).
// f32 throughout (inputs/outputs are f32; no precision to trade).
// CDNA5 paths: v_wmma_f32_16x16x4_f32 ones-matrix wave reduction,
// global_prefetch_b8 for the streaming pred walk.

typedef float v2f __attribute__((ext_vector_type(2)));
typedef float v8f __attribute__((ext_vector_type(8)));

#define BB 16   // batch rows
#define CC 32   // classes

// Wave32 all-lane sum via V_WMMA_F32_16X16X4_F32 with a ones B-matrix.
// A (16x4 f32, 2 VGPRs): lanes 0-15 hold A[m,0](v0)/A[m,1](v1), lanes 16-31
// hold A[m,2]/A[m,3]. Put acc in a[0], 0 in a[1] -> D[m,n] = acc_m + acc_{m+16}
// for every column n (B all ones makes B's lane layout irrelevant).
// Then 8 in-lane adds over D's VGPRs + one xor-16 shuffle = full wave sum.
static __device__ __forceinline__ float wave_sum_wmma(float acc) {
  v2f a; a[0] = acc;  a[1] = 0.0f;
  v2f b; b[0] = 1.0f; b[1] = 1.0f;
  v8f c = {};
  v8f d = __builtin_amdgcn_wmma_f32_16x16x4_f32(
      /*neg_a=*/false, a, /*neg_b=*/false, b,
      /*c_mod=*/(short)0, c, /*reuse_a=*/false, /*reuse_b=*/false);
  float s = ((d[0] + d[1]) + (d[2] + d[3])) + ((d[4] + d[5]) + (d[6] + d[7]));
  s += __shfl_xor(s, 16, 32);
  return s;
}

__global__ void __launch_bounds__(256) fpce_main(const float* __restrict__ pred,
                                                 const int*   __restrict__ tru,
                                                 float* __restrict__ partial,
                                                 int N) {
  const int n = blockIdx.x * 256 + threadIdx.x;   // N is a multiple of 256

  // counts[c] = #{b' : true[b',n] == c}; branchless compares, no dynamic
  // per-lane indexing (which would spill to scratch).
  int cnt[CC];
  #pragma unroll
  for (int cix = 0; cix < CC; ++cix) cnt[cix] = 0;
  #pragma unroll 1
  for (int b = 0; b < BB; ++b) {
    const int t = tru[(size_t)b * N + n];
    #pragma unroll
    for (int cix = 0; cix < CC; ++cix) cnt[cix] += (t == cix) ? 1 : 0;
  }

  float acc = 0.0f;
  const float* colbase = pred + n;                // lane-consecutive n => coalesced
  #pragma unroll 1
  for (int b = 0; b < BB; ++b) {
    const float* pb = colbase + (size_t)b * CC * N;
    if (b + 1 < BB)                                // stream-prefetch next b stripe
      __builtin_prefetch(colbase + (size_t)(b + 1) * CC * N, 0, 3);

    float x[CC];
    #pragma unroll
    for (int cix = 0; cix < CC; ++cix) x[cix] = pb[(size_t)cix * N];

    float m = x[0];
    #pragma unroll
    for (int cix = 1; cix < CC; ++cix) m = fmaxf(m, x[cix]);

    float s = 0.0f;
    #pragma unroll
    for (int cix = 0; cix < CC; ++cix) s += __expf(x[cix] - m);

    const float M = m + __logf(s);                 // log-sum-exp shift
    #pragma unroll
    for (int cix = 0; cix < CC; ++cix) {
      const float lp = x[cix] - M;                 // log p
      const float f  = -lp * __expf(-lp);          // -log(p)/p
      acc = fmaf(f, (float)cnt[cix], acc);
    }
  }

  // wave32 sum via WMMA, then 8 waves -> block partial through LDS
  const float wsum = wave_sum_wmma(acc);
  __shared__ float sbuf[8];
  const int wave = threadIdx.x >> 5;
  const int lane = threadIdx.x & 31;
  if (lane == 0) sbuf[wave] = wsum;
  __syncthreads();
  if (threadIdx.x == 0) {
    float t = 0.0f;
    #pragma unroll
    for (int w = 0; w < 8; ++w) t += sbuf[w];
    partial[blockIdx.x] = t;
  }
}

__global__ void __launch_bounds__(32) fpce_reduce(const float* __restrict__ partial,
                                                  float* __restrict__ out,
                                                  int nparts, float scale) {
  float acc = 0.0f;
  for (int i = threadIdx.x; i < nparts; i += 32) acc += partial[i];
  const float s = wave_sum_wmma(acc);
  if (threadIdx.x == 0) out[0] = s * scale;       // d_out fully overwritten
}

extern "C" void kernel_launch(void* const* d_in, const int* in_sizes, int n_in,
                              void* d_out, int out_size, void* d_ws, size_t ws_size,
                              hipStream_t stream) {
  const float* pred = (const float*)d_in[0];   // (16, 32, N) f32
  const int*   tru  = (const int*)d_in[1];     // (16, N) i32
  float* partial = (float*)d_ws;
  float* out     = (float*)d_out;

  const int N = in_sizes[0] / (BB * CC);       // 131072
  const int blocks = N / 256;                  // 512 (N multiple of 256)

  fpce_main<<<blocks, 256, 0, stream>>>(pred, tru, partial, N);
  fpce_reduce<<<1, 32, 0, stream>>>(partial, out, blocks, 1.0f / (float)N);
}